// e3jLayer_23398981829381
// MI455X (gfx1250) — compile-verified
//
#include <hip/hip_runtime.h>

// ---------------------------------------------------------------------------
// e3nn-style l<=1 equivariant message passing layer for MI455X (gfx1250).
//
// Roofline: working set (node_feats 25.6MB + agg 25.6MB + pos 0.6MB) fits the
// 192MB L2, so the edge pass is L2-bandwidth / gather-latency bound, not HBM
// (23.3 TB/s) bound, and compute (~2 GFLOP) is free -> keep fp32 end-to-end.
//
// One wave32 per edge:
//   - per-edge 4x4 coupling matrix M[i][k] = sum_j sh[j]*CG[i][j][k]
//   - tp(32x4) = sf(32x4) @ M(4x4) as two V_WMMA_F32_16X16X4_F32
//     (A = 16 feature rows x K=4 irreps, B = M in columns 0..3, rest zero)
//   - both A-half loads issued before the WMMAs (pipelined), single merged
//     scatter region of 16 fp32 global atomics (agent scope -> resolved in L2)
// Then a small GEMV kernel applies w_linear / 16.
// ---------------------------------------------------------------------------

#define F_DIM 32
#define IRREP 4

typedef float v2f __attribute__((ext_vector_type(2)));
typedef float v8f __attribute__((ext_vector_type(8)));

// Precomputed real Clebsch-Gordan coupling tensor CG[i][j][k] for MAX_L=1
// (index order: 0 = l=0 scalar; 1,2,3 = l=1 (y,z,x) as in the reference SH).
// Block structure from _build_cg: 0x0->0 identity, 0x1->1 / 1x0->1 identity,
// 1x1->0 = delta/sqrt(3), 1x1->1 = epsilon/sqrt(2).
#define C110 0.57735026918962576f /* 1/sqrt(3) */
#define C111 0.70710678118654752f /* 1/sqrt(2) */
__constant__ float c_CG[IRREP][IRREP][IRREP] = {
    // i = 0 (scalar)
    { {1.f, 0.f, 0.f, 0.f},
      {0.f, 1.f, 0.f, 0.f},
      {0.f, 0.f, 1.f, 0.f},
      {0.f, 0.f, 0.f, 1.f} },
    // i = 1 (y)
    { {0.f, 1.f, 0.f, 0.f},
      {C110, 0.f, 0.f, 0.f},
      {0.f, 0.f, 0.f,  C111},
      {0.f, 0.f, -C111, 0.f} },
    // i = 2 (z)
    { {0.f, 0.f, 1.f, 0.f},
      {0.f, 0.f, 0.f, -C111},
      {C110, 0.f, 0.f, 0.f},
      {0.f,  C111, 0.f, 0.f} },
    // i = 3 (x)
    { {0.f, 0.f, 0.f, 1.f},
      {0.f, 0.f,  C111, 0.f},
      {0.f, -C111, 0.f, 0.f},
      {C110, 0.f, 0.f, 0.f} },
};

// One wave32 per edge. blockDim.x = 256 (8 waves / 8 edges per block).
__global__ __launch_bounds__(256) void e3j_edge_tp_wmma_kernel(
    const float* __restrict__ pos,       // [N,3]
    const float* __restrict__ nfeat,     // [N,F,4]
    const int*   __restrict__ senders,   // [E]
    const int*   __restrict__ receivers, // [E]
    float*       __restrict__ agg,       // [N,F,4] accumulator (pre-zeroed)
    int E)
{
    const int lane = threadIdx.x & 31;
    const int e    = blockIdx.x * 8 + (threadIdx.x >> 5);
    if (e >= E) return;   // wave-uniform: surviving waves keep EXEC all-ones (WMMA requirement)

    const int s = senders[e];
    const int r = receivers[e];

    // Hide gather latency: prefetch a future edge's sender feature row into
    // cache (lowers to global_prefetch_b8 on gfx1250).
    {
        const int ep = e + 64;
        if (ep < E) {
            __builtin_prefetch(&nfeat[(size_t)senders[ep] * (F_DIM * IRREP)], 0, 1);
        }
    }

    // ---- A matrix loads first (both feature halves), so their latency is
    // hidden under the sh / B-matrix VALU work and the first WMMA. ----
    //   A (16x4 fp32, 2 VGPRs/lane): VGPR0: lanes 0-15 -> K=0, 16-31 -> K=2;
    //                                VGPR1: K=1 / K=3. Rows M = feature index.
    const int    fr    = lane & 15;
    const int    kbase = (lane < 16) ? 0 : 2;
    const size_t srow  = (size_t)s * (F_DIM * IRREP);
    const float* ap0 = &nfeat[srow + (size_t)fr * IRREP + kbase];          // f 0..15
    const float* ap1 = &nfeat[srow + (size_t)(16 + fr) * IRREP + kbase];   // f 16..31
    v2f a0; a0.x = ap0[0]; a0.y = ap0[1];   // contiguous -> global_load_b64
    v2f a1; a1.x = ap1[0]; a1.y = ap1[1];

    // Edge direction + real spherical harmonics (l<=1), order [c0, y, z, x].
    float dx = pos[3 * r + 0] - pos[3 * s + 0];
    float dy = pos[3 * r + 1] - pos[3 * s + 1];
    float dz = pos[3 * r + 2] - pos[3 * s + 2];
    const float inv = __frsqrt_rn(dx * dx + dy * dy + dz * dz);
    dx *= inv; dy *= inv; dz *= inv;
    const float kSH1 = 0.4886025119029199f;
    const float sh[4] = {0.28209479177387814f, kSH1 * dy, kSH1 * dz, kSH1 * dx};

    // ---- B matrix (4x16 fp32, 2 VGPRs/lane) ----
    // CDNA5 B layout (row striped across lanes within a VGPR, rows split like
    // A's K halves): VGPR0: lanes 0-15 -> row K=0, lanes 16-31 -> row K=2;
    //                VGPR1: lanes 0-15 -> row K=1, lanes 16-31 -> row K=3.
    // Column n = lane&15. Only columns 0..3 carry M; 4..15 are zero padding.
    const int n  = lane & 15;
    const int i0 = (lane < 16) ? 0 : 2;
    float m0 = 0.f, m1 = 0.f;
    if (n < IRREP) {
#pragma unroll
        for (int j = 0; j < 4; ++j) {
            m0 += sh[j] * c_CG[i0][j][n];
            m1 += sh[j] * c_CG[i0 + 1][j][n];
        }
    }
    v2f b; b.x = m0; b.y = m1;

    // ---- Two independent WMMAs (f 0..15 and f 16..31), no D->A/B hazard ----
    v8f acc0 = {};
    v8f d0 = __builtin_amdgcn_wmma_f32_16x16x4_f32(
        false, a0, false, b, (short)0, acc0, false, false);
    v8f acc1 = {};
    v8f d1 = __builtin_amdgcn_wmma_f32_16x16x4_f32(
        false, a1, false, b, (short)0, acc1, false, false);

    // ---- Merged scatter: D VGPR q holds feature row (q | q+8) per lane half;
    // lane with column n<4 owns irrep component n. 16 fp32 atomics, one EXEC
    // masked region -> clause-batched global_atomic_add_f32 (DEV scope / L2).
    if (n < IRREP) {
        const int fhalf_lo = (lane < 16) ? 0 : 8;
        float* base0 = &agg[((size_t)r * F_DIM + (size_t)fhalf_lo) * IRREP + n];
        float* base1 = base0 + (size_t)16 * IRREP;
#pragma unroll
        for (int q = 0; q < 8; ++q) {
            __hip_atomic_fetch_add(base0 + (size_t)q * IRREP, d0[q],
                                   __ATOMIC_RELAXED, __HIP_MEMORY_SCOPE_AGENT);
        }
#pragma unroll
        for (int q = 0; q < 8; ++q) {
            __hip_atomic_fetch_add(base1 + (size_t)q * IRREP, d1[q],
                                   __ATOMIC_RELAXED, __HIP_MEMORY_SCOPE_AGENT);
        }
    }
}

// out[n,f] = (sum_k agg[n,f,k] * w[k]) / 16
__global__ __launch_bounds__(256) void e3j_linear_kernel(
    const float* __restrict__ agg, const float* __restrict__ w,
    float* __restrict__ out, int total /* N*F */)
{
    const int idx = blockIdx.x * blockDim.x + threadIdx.x;
    if (idx >= total) return;
    const float4 v = ((const float4*)agg)[idx];
    out[idx] = (v.x * w[0] + v.y * w[1] + v.z * w[2] + v.w * w[3]) * 0.0625f;
}

extern "C" void kernel_launch(void* const* d_in, const int* in_sizes, int n_in,
                              void* d_out, int out_size, void* d_ws, size_t ws_size,
                              hipStream_t stream) {
    const float* pos   = (const float*)d_in[0];  // [N,3]
    const float* nfeat = (const float*)d_in[1];  // [N,32,4]
    const float* w     = (const float*)d_in[2];  // [4,1]
    const int*   snd   = (const int*)d_in[3];    // [E]
    const int*   rcv   = (const int*)d_in[4];    // [E]

    const int E     = in_sizes[3];
    const int total = in_sizes[1] / IRREP;       // N*F

    float* agg = (float*)d_ws;                   // [N,F,4] fp32 scratch
    (void)hipMemsetAsync(agg, 0, (size_t)total * IRREP * sizeof(float), stream);

    const int edge_blocks = (E + 7) / 8;         // 8 edges (waves) per 256-thread block
    e3j_edge_tp_wmma_kernel<<<edge_blocks, 256, 0, stream>>>(pos, nfeat, snd, rcv, agg, E);

    e3j_linear_kernel<<<(total + 255) / 256, 256, 0, stream>>>(agg, w, (float*)d_out, total);
}